// TieredGroupAttentionTier_39848706573738
// MI455X (gfx1250) — compile-verified
//
#include <hip/hip_runtime.h>

// ---------------------------------------------------------------------------
// TieredGroupAttention on MI455X (gfx1250), linear-attention reassociation.
// No softmax in the reference => (qK^T)v == q(K^T v). Total ~40 GFLOP,
// ~135 MB traffic (L2-resident on 192MB L2; HBM time ~us at 23.3 TB/s).
// All GEMMs via v_wmma_f32_16x16x4_f32; global->LDS staging uses gfx1250
// GLOBAL_LOAD_ASYNC_TO_LDS_* with double-buffered LDS + s_wait_asynccnt.
// ---------------------------------------------------------------------------

typedef __attribute__((ext_vector_type(2))) float v2f;
typedef __attribute__((ext_vector_type(8))) float v8f;

#define TILE_M 128
#define TILE_N 128
#define TILE_K 16
#define LDA_PAD 20 // 80B rows: 16B-aligned for b128 async, conflict-spread b64 reads
#define LDB_PAD 18 // 72B rows: 8B-aligned v2f reads, conflict-spread

#if defined(__gfx1250__) &&                                             \
    __has_builtin(__builtin_amdgcn_global_load_async_to_lds_b128) &&    \
    __has_builtin(__builtin_amdgcn_global_load_async_to_lds_b32) &&     \
    __has_builtin(__builtin_amdgcn_s_wait_asynccnt)
#define ASYNC_COPY 1
#else
#define ASYNC_COPY 0
#endif

#if ASYNC_COPY
// Param types per hipcc diagnostics: b128 takes 'int __attribute__((
// vector_size(16))) __device__*' (AS1 = global) first; LDS side is AS3.
typedef int async_b128_t __attribute__((vector_size(16)));
typedef __attribute__((address_space(1))) async_b128_t* g128p;
typedef __attribute__((address_space(3))) async_b128_t* l128p;
typedef __attribute__((address_space(1))) int* g32p;
typedef __attribute__((address_space(3))) int* l32p;
#define GLB2LDS_B128(SRC, DST)                                          \
  __builtin_amdgcn_global_load_async_to_lds_b128(                       \
      (g128p)(void*)(SRC), (l128p)(void*)(DST), 0, 0)
#define GLB2LDS_B32(SRC, DST)                                           \
  __builtin_amdgcn_global_load_async_to_lds_b32(                        \
      (g32p)(void*)(SRC), (l32p)(void*)(DST), 0, 0)
#define WAIT_ASYNC(N) __builtin_amdgcn_s_wait_asynccnt(N)
#else
#define WAIT_ASYNC(N) ((void)0)
#endif

// Stage one K-tile of A (as[m][k]) and B (bs[n][k], transposed) into LDS.
// Issue counts are constant (guards are address clamps, not branches) so the
// per-wave ASYNCcnt bookkeeping is exact. Clamped rows/cols only feed C
// elements that the guarded store never writes.
template <int TRANSA>
__device__ __forceinline__ void stage_tile(
    const float* __restrict__ A, const float* __restrict__ B, int k0,
    float (*__restrict__ as)[LDA_PAD], float (*__restrict__ bs)[LDB_PAD],
    int tileM, int tileN, int M, int N, int lda, int ldb, int t) {
  // ---- A tile ----
  if (TRANSA == 0) {
    const int r = t >> 1;       // 0..127
    const int kq = (t & 1) * 8; // 0 or 8
    int gm = tileM + r;
    gm = gm < M ? gm : (M - 1);
    const float* src = A + (long long)gm * lda + k0 + kq;
#if ASYNC_COPY
    GLB2LDS_B128(src, &as[r][kq]);
    GLB2LDS_B128(src + 4, &as[r][kq + 4]);
#else
    const float4 p0 = *(const float4*)(src);
    const float4 p1 = *(const float4*)(src + 4);
    as[r][kq + 0] = p0.x; as[r][kq + 1] = p0.y;
    as[r][kq + 2] = p0.z; as[r][kq + 3] = p0.w;
    as[r][kq + 4] = p1.x; as[r][kq + 5] = p1.y;
    as[r][kq + 6] = p1.z; as[r][kq + 7] = p1.w;
#endif
  } else { // A stored [K,M]: transpose-scatter into as[m][k]
    const int kk = t >> 4;       // 0..15
    const int mq = (t & 15) * 8; // 0..120
    const float* rowp = A + (long long)(k0 + kk) * lda;
#pragma unroll
    for (int i = 0; i < 8; ++i) {
      int gm = tileM + mq + i;
      gm = gm < M ? gm : (M - 1);
#if ASYNC_COPY
      GLB2LDS_B32(rowp + gm, &as[mq + i][kk]);
#else
      as[mq + i][kk] = rowp[gm];
#endif
    }
  }
  // ---- B tile: transpose-scatter global [K,N] row -> bs[n][k] ----
  {
    const int kk = t >> 4;       // 0..15
    const int nq = (t & 15) * 8; // 0..120
    const float* rowp = B + (long long)(k0 + kk) * ldb;
#pragma unroll
    for (int i = 0; i < 8; ++i) {
      int gn = tileN + nq + i;
      gn = gn < N ? gn : (N - 1);
#if ASYNC_COPY
      GLB2LDS_B32(rowp + gn, &bs[nq + i][kk]);
#else
      bs[nq + i][kk] = rowp[gn];
#endif
    }
  }
}

// Generic batched GEMM:  C[b] = alpha * op(A[b]) @ B[b] (+ bias row)
//   TRANSA=0: A is [M,K] row-major.  TRANSA=1: A stored [K,M] (C = A^T B).
// Batch offsets: Aoff=b*aS1; Boff=(b/bDiv)*bS1; Coff=(b/cDiv)*cS1+(b%cDiv)*cS2
// Requires K % 16 == 0 (all call sites satisfy this).
template <int TRANSA>
__global__ __launch_bounds__(256) void gemm_f32_wmma(
    const float* __restrict__ A0, long long aS1,
    const float* __restrict__ B0, long long bS1, int bDiv,
    const float* __restrict__ bias0, long long biasS, int hasBias,
    float* __restrict__ C0, long long cS1, long long cS2, int cDiv,
    int M, int N, int K, int lda, int ldb, int ldc, float alpha) {
  __shared__ float As[2][TILE_M][LDA_PAD];
  __shared__ float Bs[2][TILE_N][LDB_PAD];

  const int b = blockIdx.z;
  const int tileM = blockIdx.y * TILE_M;
  const int tileN = blockIdx.x * TILE_N;

  const float* __restrict__ A = A0 + (long long)b * aS1;
  const float* __restrict__ B = B0 + (long long)(b / bDiv) * bS1;
  float* __restrict__ C =
      C0 + (long long)(b / cDiv) * cS1 + (long long)(b % cDiv) * cS2;

  const int t = threadIdx.x;
  const int lane = t & 31;
  const int wid = t >> 5;  // 8 wave32 waves
  const int wm = wid & 3;  // wave row group: 4 x 32 rows
  const int wn = wid >> 2; // wave col group: 2 x 64 cols
  const int lm = lane & 15;
  const int lh = lane >> 4; // lane half selects K pair (K0/K1 vs K2/K3)

  // async issue count per stage per thread (must match stage_tile exactly)
  constexpr int NPER = (TRANSA == 0) ? 10 : 16;

  const v8f vzero = {0.f, 0.f, 0.f, 0.f, 0.f, 0.f, 0.f, 0.f};
  v8f acc[2][4];
#pragma unroll
  for (int i = 0; i < 2; ++i)
#pragma unroll
    for (int j = 0; j < 4; ++j) acc[i][j] = vzero;

  const int nkt = K / TILE_K;
  stage_tile<TRANSA>(A, B, 0, As[0], Bs[0], tileM, tileN, M, N, lda, ldb, t);

  for (int kt = 0; kt < nkt; ++kt) {
    const int cur = kt & 1;
    if (kt + 1 < nkt) {
      // prefetch next tile into the other buffer (async engine fills LDS)
      stage_tile<TRANSA>(A, B, (kt + 1) * TILE_K, As[cur ^ 1], Bs[cur ^ 1],
                         tileM, tileN, M, N, lda, ldb, t);
      WAIT_ASYNC(NPER); // drain current stage; leave next stage in flight
    } else {
      WAIT_ASYNC(0);
    }
    __syncthreads(); // current tile globally visible

    float (*__restrict__ as)[LDA_PAD] = As[cur];
    float (*__restrict__ bs)[LDB_PAD] = Bs[cur];
#pragma unroll
    for (int ks = 0; ks < TILE_K; ks += 4) {
      const int kc = ks + lh * 2; // lanes 16-31 hold K2/K3 per ISA A layout
      v2f af[2];
      v2f bf[4];
#pragma unroll
      for (int i = 0; i < 2; ++i)
        af[i] = *(const v2f*)&as[wm * 32 + i * 16 + lm][kc];
#pragma unroll
      for (int j = 0; j < 4; ++j)
        bf[j] = *(const v2f*)&bs[wn * 64 + j * 16 + lm][kc];
#pragma unroll
      for (int i = 0; i < 2; ++i)
#pragma unroll
        for (int j = 0; j < 4; ++j)
          acc[i][j] = __builtin_amdgcn_wmma_f32_16x16x4_f32(
              false, af[i], false, bf[j], (short)0, acc[i][j], false, false);
    }
    __syncthreads(); // safe to overwrite this buffer next iteration
  }

  // ---- store C (16x16 f32 D layout: VGPR v -> M = v (+8 for hi lanes)) ----
#pragma unroll
  for (int j = 0; j < 4; ++j) {
    const int n = tileN + wn * 64 + j * 16 + lm;
    float bv = 0.f;
    if (hasBias && n < N) bv = bias0[(long long)b * biasS + n];
#pragma unroll
    for (int i = 0; i < 2; ++i) {
      const int m0 = tileM + wm * 32 + i * 16 + lh * 8;
#pragma unroll
      for (int v = 0; v < 8; ++v) {
        const int m = m0 + v;
        if (m < M && n < N)
          C[(long long)m * ldc + n] = alpha * acc[i][j][v] + bv;
      }
    }
  }
}

// tiny batched vec-mat:  out[b][n] = alpha * sum_k vec[b][k]*Mat[b/mDiv][k][n] (+bias)
__global__ __launch_bounds__(256) void vecmat_f32(
    const float* __restrict__ v0, long long vS,
    const float* __restrict__ m0, long long mS, int mDiv, int ldm,
    const float* __restrict__ bias0, long long biasS, int hasBias,
    float* __restrict__ o0, long long oS, int K, int Nd, int totalB,
    float alpha) {
  const int idx = blockIdx.x * 256 + threadIdx.x;
  const int b = idx / Nd;
  const int n = idx % Nd;
  if (b >= totalB) return;
  const float* v = v0 + (long long)b * vS;
  const float* m = m0 + (long long)(b / mDiv) * mS;
  float s = 0.f;
  for (int k = 0; k < K; ++k) s += v[k] * m[(long long)k * ldm + n];
  float bv = hasBias ? bias0[(long long)b * biasS + n] : 0.f;
  o0[(long long)b * oS + n] = alpha * s + bv;
}

extern "C" void kernel_launch(void* const* d_in, const int* in_sizes, int n_in,
                              void* d_out, int out_size, void* d_ws,
                              size_t ws_size, hipStream_t stream) {
  (void)in_sizes; (void)n_in; (void)out_size; (void)ws_size;
  const int Ntok = 2048, Dd = 512, Aa = 64, NQc = 8, G1c = 8, G2c = 8,
            A1c = 128;
  const float scale1 = 0.08838834764831845f; // 1/sqrt(128)
  const float scale2 = 0.125f;               // 1/sqrt(64)

  const float* x   = (const float*)d_in[0];
  const float* Wq1 = (const float*)d_in[1];
  const float* bq1 = (const float*)d_in[2];
  const float* Wk1 = (const float*)d_in[3];
  const float* bk1 = (const float*)d_in[4];
  const float* Wv1 = (const float*)d_in[5];
  const float* bv1 = (const float*)d_in[6];
  const float* Wo1 = (const float*)d_in[7];
  const float* bo1 = (const float*)d_in[8];
  const float* Wq2 = (const float*)d_in[9];
  const float* bq2 = (const float*)d_in[10];
  const float* Wk2 = (const float*)d_in[11];
  const float* bk2 = (const float*)d_in[12];
  const float* Wv2 = (const float*)d_in[13];
  const float* bv2 = (const float*)d_in[14];
  const float* Wo2 = (const float*)d_in[15];
  const float* bo2 = (const float*)d_in[16];
  const float* Wo  = (const float*)d_in[17];
  const float* bo  = (const float*)d_in[18];
  float* outP = (float*)d_out;

  float* ws = (float*)d_ws;
  size_t off = 0;
  auto alloc = [&](size_t n) {
    float* p = ws + off;
    off += (n + 63) & ~(size_t)63;
    return p;
  };
  float* k1    = alloc((size_t)G1c * Ntok * A1c);
  float* v1    = alloc((size_t)G1c * Ntok * A1c);
  float* KV1   = alloc((size_t)G1c * A1c * A1c);
  float* bqkv1 = alloc((size_t)G1c * NQc * A1c);
  float* T1    = alloc((size_t)G1c * Dd * NQc * A1c);
  float* Weff1 = alloc((size_t)G1c * Dd * Dd);
  float* beff1 = alloc((size_t)G1c * Dd);
  float* o1    = alloc((size_t)G1c * Ntok * Dd);
  float* k2    = alloc((size_t)G2c * Ntok * Aa);
  float* v2    = alloc((size_t)G2c * Ntok * Aa);
  float* KV2   = alloc((size_t)G2c * Aa * Aa);
  float* bqkv2 = alloc((size_t)G2c * NQc * Aa);
  float* T2    = alloc((size_t)G2c * Dd * NQc * Aa);
  float* Weff2 = alloc((size_t)G2c * Dd * Dd);
  float* beff2 = alloc((size_t)G2c * Dd);
  float* o2    = alloc((size_t)G2c * Ntok * Dd);

  auto gemm = [&](bool transA, const float* A, long long aS1, const float* B,
                  long long bS1, int bDiv, const float* bias, long long biasS,
                  int hasBias, float* Cm, long long cS1, long long cS2,
                  int cDiv, int M, int Nn, int K, int lda, int ldb, int ldc,
                  int batches, float alpha) {
    dim3 grid((Nn + TILE_N - 1) / TILE_N, (M + TILE_M - 1) / TILE_M, batches);
    if (transA)
      gemm_f32_wmma<1><<<grid, 256, 0, stream>>>(
          A, aS1, B, bS1, bDiv, bias, biasS, hasBias, Cm, cS1, cS2, cDiv, M,
          Nn, K, lda, ldb, ldc, alpha);
    else
      gemm_f32_wmma<0><<<grid, 256, 0, stream>>>(
          A, aS1, B, bS1, bDiv, bias, biasS, hasBias, Cm, cS1, cS2, cDiv, M,
          Nn, K, lda, ldb, ldc, alpha);
  };

  // ---------------- Tier 1 ----------------
  // k1 = x @ Wk1 + bk1 ; v1 = x @ Wv1 + bv1        [g, N, A1]
  gemm(false, x, 0, Wk1, (long long)Dd * A1c, 1, bk1, A1c, 1, k1,
       (long long)Ntok * A1c, 0, 1, Ntok, A1c, Dd, Dd, A1c, A1c, G1c, 1.f);
  gemm(false, x, 0, Wv1, (long long)Dd * A1c, 1, bv1, A1c, 1, v1,
       (long long)Ntok * A1c, 0, 1, Ntok, A1c, Dd, Dd, A1c, A1c, G1c, 1.f);
  // KV1 = k1^T @ v1                                [g, A1, A1]
  gemm(true, k1, (long long)Ntok * A1c, v1, (long long)Ntok * A1c, 1, nullptr,
       0, 0, KV1, (long long)A1c * A1c, 0, 1, A1c, A1c, Ntok, A1c, A1c, A1c,
       G1c, 1.f);
  // bqkv1[g,qi,:] = bq1 @ KV1 * scale1             [g, NQ*A1]
  vecmat_f32<<<(G1c * NQc * A1c + 255) / 256, 256, 0, stream>>>(
      bq1, A1c, KV1, (long long)A1c * A1c, NQc, A1c, nullptr, 0, 0, bqkv1,
      A1c, A1c, A1c, G1c * NQc, scale1);
  // T1[g][:, qi*A1: ] = Wq1[g,qi] @ KV1[g] * scale1   [g, D, NQ*A1]
  gemm(false, Wq1, (long long)Dd * A1c, KV1, (long long)A1c * A1c, NQc,
       nullptr, 0, 0, T1, (long long)Dd * NQc * A1c, A1c, NQc, Dd, A1c, A1c,
       A1c, A1c, NQc * A1c, G1c * NQc, scale1);
  // Weff1 = T1 @ Wo1                               [g, D, D]
  gemm(false, T1, (long long)Dd * NQc * A1c, Wo1, (long long)NQc * A1c * Dd,
       1, nullptr, 0, 0, Weff1, (long long)Dd * Dd, 0, 1, Dd, Dd, NQc * A1c,
       NQc * A1c, Dd, Dd, G1c, 1.f);
  // beff1 = bqkv1 @ Wo1 + bo1                      [g, D]
  vecmat_f32<<<(G1c * Dd + 255) / 256, 256, 0, stream>>>(
      bqkv1, (long long)NQc * A1c, Wo1, (long long)NQc * A1c * Dd, 1, Dd, bo1,
      Dd, 1, beff1, Dd, NQc * A1c, Dd, G1c, 1.f);
  // o1 = x @ Weff1 + beff1                         [g, N, D]
  gemm(false, x, 0, Weff1, (long long)Dd * Dd, 1, beff1, Dd, 1, o1,
       (long long)Ntok * Dd, 0, 1, Ntok, Dd, Dd, Dd, Dd, Dd, G1c, 1.f);

  // ---------------- Tier 2 ----------------
  gemm(false, o1, (long long)Ntok * Dd, Wk2, (long long)Dd * Aa, 1, bk2, Aa,
       1, k2, (long long)Ntok * Aa, 0, 1, Ntok, Aa, Dd, Dd, Aa, Aa, G2c, 1.f);
  gemm(false, o1, (long long)Ntok * Dd, Wv2, (long long)Dd * Aa, 1, bv2, Aa,
       1, v2, (long long)Ntok * Aa, 0, 1, Ntok, Aa, Dd, Dd, Aa, Aa, G2c, 1.f);
  gemm(true, k2, (long long)Ntok * Aa, v2, (long long)Ntok * Aa, 1, nullptr,
       0, 0, KV2, (long long)Aa * Aa, 0, 1, Aa, Aa, Ntok, Aa, Aa, Aa, G2c,
       1.f);
  vecmat_f32<<<(G2c * NQc * Aa + 255) / 256, 256, 0, stream>>>(
      bq2, Aa, KV2, (long long)Aa * Aa, NQc, Aa, nullptr, 0, 0, bqkv2, Aa, Aa,
      Aa, G2c * NQc, scale2);
  gemm(false, Wq2, (long long)Dd * Aa, KV2, (long long)Aa * Aa, NQc, nullptr,
       0, 0, T2, (long long)Dd * NQc * Aa, Aa, NQc, Dd, Aa, Aa, Aa, Aa,
       NQc * Aa, G2c * NQc, scale2);
  gemm(false, T2, (long long)Dd * NQc * Aa, Wo2, (long long)NQc * Aa * Dd, 1,
       nullptr, 0, 0, Weff2, (long long)Dd * Dd, 0, 1, Dd, Dd, NQc * Aa,
       NQc * Aa, Dd, Dd, G2c, 1.f);
  vecmat_f32<<<(G2c * Dd + 255) / 256, 256, 0, stream>>>(
      bqkv2, (long long)NQc * Aa, Wo2, (long long)NQc * Aa * Dd, 1, Dd, bo2,
      Dd, 1, beff2, Dd, NQc * Aa, Dd, G2c, 1.f);
  gemm(false, o1, (long long)Ntok * Dd, Weff2, (long long)Dd * Dd, 1, beff2,
       Dd, 1, o2, (long long)Ntok * Dd, 0, 1, Ntok, Dd, Dd, Dd, Dd, Dd, G2c,
       1.f);

  // ---------------- Final projection ----------------
  // out = o2_flat @ Wo + bo                        [G2*N, D]
  gemm(false, o2, 0, Wo, 0, 1, bo, 0, 1, outP, 0, 0, 1, G2c * Ntok, Dd, Dd,
       Dd, Dd, Dd, 1, 1.f);
}